// DNM_Linear_M_47167330845216
// MI455X (gfx1250) — compile-verified
//
#include <hip/hip_runtime.h>

// DNM (dendritic neuron model) forward for MI455X / gfx1250.
// B=128, OUT=256, M=8, IN=512. All fp32.
//
//   log2 d = -sum_i log2(1 + exp2(z')),  z' = -(x*w+q)*log2e = x*w' + q'
//   out[b,o] = sum_m sigmoid(d*Wd[m] - Qd[m]*512)
//
// - 512-term reduction on the matrix pipe: V_WMMA_F32_16X16X4_F32, B = ones.
// - x tile (16x512) + W block (8x512, contiguous) staged via the Tensor Data
//   Mover (tensor_load_to_lds, TENSORcnt); x descriptor uses TDM LDS padding
//   (2 DWORDs per 256) -> row stride 516 floats, bank-conflict-free reads.
// - per element: 1 v_fma + 1 v_exp_f32 + 1 v_add + 1 v_log_f32 (TRANS-bound).

typedef float    v2f __attribute__((ext_vector_type(2)));
typedef float    v8f __attribute__((ext_vector_type(8)));
typedef uint32_t v4u __attribute__((ext_vector_type(4)));
typedef int      v4i __attribute__((ext_vector_type(4)));
typedef int      v8i __attribute__((ext_vector_type(8)));

#define LOG2E 1.4426950408889634f

#if __has_include(<hip/amd_detail/amd_gfx1250_TDM.h>)
// therock-10.0 headers -> clang-23 6-arg builtin
#define TDM_LOAD(g0, g1, g2, g3) \
    __builtin_amdgcn_tensor_load_to_lds((g0), (g1), (g2), (g3), (v8i){0,0,0,0,0,0,0,0}, 0)
#else
// ROCm 7.2 / clang-22 5-arg builtin: (v4u g0, v8i g1, v4i g2, v4i g3, i32 cpol)
#define TDM_LOAD(g0, g1, g2, g3) \
    __builtin_amdgcn_tensor_load_to_lds((g0), (g1), (g2), (g3), 0)
#endif

__device__ __forceinline__ uint32_t lds_offset(const void* p) {
    // generic (flat) shared pointer: low 32 bits are the LDS byte offset
    return (uint32_t)(uintptr_t)p;
}

__global__ __launch_bounds__(256)
void dnm_forward_kernel(const float* __restrict__ x,      // [128,512]
                        const float* __restrict__ Wsyn,   // [256,8,512]
                        const float* __restrict__ Qsyn,   // [256,8,512]
                        const float* __restrict__ Wd,     // [8]
                        const float* __restrict__ Qd,     // [8]
                        float* __restrict__ out)          // [128,256]
{
    // x tile: 16 rows x 512, TDM-padded to row stride 516 floats
    // (pad 2 DWORDs every 256 DWORDs) -> lane t reads banks {4t, 4t+1}
    // (kh=0) / {4t+2, 4t+3} (kh=1): all 32 lanes conflict-free.
    __shared__ float xs[16 * 516];
    __shared__ float ws[8 * 512];     // W rows for this block's 8 dendrites
    __shared__ float red[8][16];

    const int tid = threadIdx.x;
    const int b0  = blockIdx.x * 16;   // 8 b-tiles
    const int o   = blockIdx.y;        // 256 outputs

    // ---- TDM: async DMA of x tile (32KB, padded) and W block (16KB) ----
    if (tid < 32) {
        const uint64_t gax = (uint64_t)(uintptr_t)(x + (size_t)b0 * 512);
        const uint64_t gaw = (uint64_t)(uintptr_t)(Wsyn + (size_t)o * 4096);

        // D# group0: count=1 | lds_addr | global_addr(57b) | type=2
        v4u g0x = { 1u, lds_offset(xs),
                    (uint32_t)gax, (uint32_t)(gax >> 32) | 0x80000000u };
        v4u g0w = { 1u, lds_offset(ws),
                    (uint32_t)gaw, (uint32_t)(gaw >> 32) | 0x80000000u };

        // x: 2D 512x16, data_size=4B, pad_enable, pad_interval=7 (256 DW),
        //    pad_amount=1 (2 DW)  -> effective LDS row stride 516 floats
        v8i g1x = { (int)((2u << 16) | (1u << 20) | (7u << 22) | (1u << 25)),
                    (int)(512u << 16),   // tensor_dim0 = 512
                    (int)(16u  << 16),   // tensor_dim1 = 16
                    (int)(512u << 16),   // tile_dim0   = 512
                    16,                  // tile_dim1   = 16
                    512, 0, 0 };         // tensor_dim0_stride = 512

        // W: 1D 4096 elements (8 contiguous rows), no padding
        v8i g1w = { (int)(2u << 16),
                    (int)(4096u << 16),  // tensor_dim0 = 4096
                    (int)(1u    << 16),  // tensor_dim1 = 1
                    (int)(4096u << 16),  // tile_dim0   = 4096
                    0,
                    4096, 0, 0 };

        // groups 2/3: higher dims = 1 (benign)
        v4i g2 = { 1, 1, 1, 0 };
        v4i g3 = { 0, 1 << 16, 0, 0 };

        TDM_LOAD(g0x, g1x, g2, g3);
        TDM_LOAD(g0w, g1w, g2, g3);
        __builtin_amdgcn_s_wait_tensorcnt(0);
    }
    __syncthreads();

    // ---- pre-scale W in LDS: w' = -w*log2e (moves scaling out of hot loop)
    for (int j = tid; j < 4096; j += 256)   // uniform trip count (16)
        ws[j] *= -LOG2E;
    __syncthreads();

    const int m    = __builtin_amdgcn_readfirstlane(tid >> 5); // wave = dendrite m
    const int lane = tid & 31;
    const int t    = lane & 15;    // A-matrix row (b index within tile)
    const int kh   = lane >> 4;    // K half: 0 -> K={0,1}, 1 -> K={2,3}

    const float* qrow = Qsyn + (size_t)(o * 8 + m) * 512;  // wave-uniform
    __builtin_prefetch(qrow, 0, 3);                        // global_prefetch_b8

    const float* xrow = &xs[t * 516];
    const float* wrow = &ws[m << 9];

    v8f acc = {0.f, 0.f, 0.f, 0.f, 0.f, 0.f, 0.f, 0.f};
    const v2f bones = {1.0f, 1.0f};    // B = ones(4x16) -> D cols = row sums

    for (int i0 = 0; i0 < 512; i0 += 4) {
        const int icol = i0 + (kh << 1);           // lane's two K slots
        const int xadj = (i0 >> 8) << 1;           // TDM pad fix-up (+2 past col 256)

        float2 xv = *(const float2*)(xrow + icol + xadj);  // ds_load_b64, no conflicts
        float2 wv = *(const float2*)(wrow + icol);         // ds_load_b64, broadcast

        float4 qq = *(const float4*)(qrow + i0);           // uniform -> s_load_b128
        float q0 = (kh ? qq.z : qq.x) * (-LOG2E);
        float q1 = (kh ? qq.w : qq.y) * (-LOG2E);

        float z0 = fmaf(xv.x, wv.x, q0);           // z' = -(x*w+q)*log2e
        float z1 = fmaf(xv.y, wv.y, q1);

        v2f a;
        a.x = __builtin_amdgcn_logf(1.0f + __builtin_amdgcn_exp2f(z0));
        a.y = __builtin_amdgcn_logf(1.0f + __builtin_amdgcn_exp2f(z1));

        // acc[b-row] += sum_k a[b-row, k]   (f32 matrix pipe)
        acc = __builtin_amdgcn_wmma_f32_16x16x4_f32(
                  false, a, false, bones, (short)0, acc, false, false);
    }

    // ---- extract row sums: row r -> VGPR r&7; lanes<16: M=0..7, lanes>=16: M=8..15
    float Ls = acc[0];
#pragma unroll
    for (int r = 1; r < 8; ++r)
        Ls = ((lane & 7) == r) ? (float)acc[r] : Ls;

    // d = prod_i sigmoid(.) = exp2(-Ls);  soma = sigmoid(d*Wd[m] - Qd[m]*512)
    float dprod = __builtin_amdgcn_exp2f(-Ls);
    float wd = Wd[m];
    float qd = Qd[m];
    float y  = fmaf(dprod, wd, -qd * 512.0f);
    float e  = __builtin_amdgcn_exp2f(-y * LOG2E);
    float soma = __builtin_amdgcn_rcpf(1.0f + e);

    // lanes 0-7 own rows 0-7, lanes 16-23 own rows 8-15 (others duplicate)
    if ((lane & 8) == 0) {
        int mrow = ((lane >> 4) << 3) | (lane & 7);
        red[m][mrow] = soma;
    }
    __syncthreads();

    // ---- sum over the 8 dendrites, one output column per block ----
    if (tid < 16) {
        float s = 0.0f;
#pragma unroll
        for (int j = 0; j < 8; ++j) s += red[j][tid];
        out[(size_t)(b0 + tid) * 256 + o] = s;
    }
}

extern "C" void kernel_launch(void* const* d_in, const int* in_sizes, int n_in,
                              void* d_out, int out_size, void* d_ws, size_t ws_size,
                              hipStream_t stream) {
    const float* x    = (const float*)d_in[0];
    const float* Wsyn = (const float*)d_in[1];
    const float* Qsyn = (const float*)d_in[2];
    const float* Wd   = (const float*)d_in[3];
    const float* Qd   = (const float*)d_in[4];
    float* out = (float*)d_out;

    dim3 grid(8, 256);   // 8 b-tiles x 256 outputs
    dim3 block(256);     // 8 waves: one per dendrite m
    hipLaunchKernelGGL(dnm_forward_kernel, grid, block, 0, stream,
                       x, Wsyn, Qsyn, Wd, Qd, out);
}